// EnhancedLNN_56693568307796
// MI455X (gfx1250) — compile-verified
//
#include <hip/hip_runtime.h>
#include <math.h>

// ---------------------------------------------------------------------------
// MI455X (gfx1250) forward pass for EnhancedLNN.
// Memory-bound on fp32 weight streaming (~140MB @ 23.3 TB/s) -> use
// V_WMMA_F32_16X16X4_F32 directly on fp32 data (no conversion traffic).
// Wave32: one 16x16 f32 tile per wave (8-VGPR accumulator).
// RK4 RHS fully fused (2 GEMMs + gate combine + axpy + state update) to cut
// graph-node count from ~500 to ~120 (launch overhead >> 6us HBM floor).
// ---------------------------------------------------------------------------

typedef float v2f __attribute__((ext_vector_type(2)));
typedef float v8f __attribute__((ext_vector_type(8)));

#define LN_EPS 1e-5f
#define BN_EPS 1e-5f
#define NROWS  256          // batch, fixed
#define ODE_STEPS 24        // fixed-step RK4 over t in [0,3]

__device__ __forceinline__ float gelu_exact(float v) {
  return 0.5f * v * (1.0f + erff(v * 0.70710678118654752f));
}
__device__ __forceinline__ float sigm(float v) {
  return 1.0f / (1.0f + expf(-v));
}

#define WMMA_F32(acc, av, bv) \
  __builtin_amdgcn_wmma_f32_16x16x4_f32(false, av, false, bv, (short)0, acc, \
                                        false, false)

// ---------------------------------------------------------------------------
// Generic GEMM: Y[256, N] = epi( X[256, K] * W[N, K]^T + bias ), ldY stride.
// EPI: 0 = none, 1 = gelu, 2 = sigmoid, 3 = batchnorm(running stats)+gelu.
// K-permuted fragment fill: within one V_WMMA_F32_16X16X4_F32 the K-set is
// {k,k+1} (lanes 0-15) U {k+4,k+5} (lanes 16-31); A and B agree, and the
// reduction is order-independent, so both fragments come from b128 loads.
// ---------------------------------------------------------------------------
template <int EPI>
__global__ __launch_bounds__(256)
void gemm_xwt_kernel(const float* __restrict__ X, const float* __restrict__ W,
                     const float* __restrict__ bias, float* __restrict__ Y,
                     int K, int N, int ldY,
                     const float* __restrict__ bn_g, const float* __restrict__ bn_b,
                     const float* __restrict__ bn_m, const float* __restrict__ bn_v) {
  const int lane = threadIdx.x & 31;
  const int wave = threadIdx.x >> 5;
  const int l    = lane & 15;
  const int hf   = lane >> 4;                     // half-wave id
  const int r0   = blockIdx.y * 16;               // row tile (M)
  const int c0   = blockIdx.x * 128 + wave * 16;  // col tile (N)
  if (c0 >= N) return;

  const float* xp = X + (size_t)(r0 + l) * K + 4 * hf;
  const float* wp = W + (size_t)(c0 + l) * K + 4 * hf;

  v8f acc = {};
  for (int k = 0; k < K; k += 16) {
    const float4 a0 = *(const float4*)(xp + k);
    const float4 w0 = *(const float4*)(wp + k);
    const float4 a1 = *(const float4*)(xp + k + 8);
    const float4 w1 = *(const float4*)(wp + k + 8);
    // speculative prefetch: OOB translation failures are silently dropped
    __builtin_prefetch(xp + k + 256, 0, 0);
    __builtin_prefetch(wp + k + 256, 0, 0);
    v2f av, bv;
    av.x = a0.x; av.y = a0.y; bv.x = w0.x; bv.y = w0.y;
    acc = WMMA_F32(acc, av, bv);
    av.x = a0.z; av.y = a0.w; bv.x = w0.z; bv.y = w0.w;
    acc = WMMA_F32(acc, av, bv);
    av.x = a1.x; av.y = a1.y; bv.x = w1.x; bv.y = w1.y;
    acc = WMMA_F32(acc, av, bv);
    av.x = a1.z; av.y = a1.w; bv.x = w1.z; bv.y = w1.w;
    acc = WMMA_F32(acc, av, bv);
  }

  const int col = c0 + l;
  const float bc = bias ? bias[col] : 0.0f;
  float bng = 0.f, bnb = 0.f, bnm = 0.f, bnr = 1.f;
  if (EPI == 3) {
    bng = bn_g[col]; bnb = bn_b[col]; bnm = bn_m[col];
    bnr = rsqrtf(bn_v[col] + BN_EPS);
  }
#pragma unroll
  for (int v = 0; v < 8; ++v) {
    // C/D layout: VGPR v holds M = v (lanes 0-15) / M = v+8 (lanes 16-31), N = l
    const int row = r0 + v + 8 * hf;
    float val = acc[v] + bc;
    if (EPI == 1) val = gelu_exact(val);
    if (EPI == 2) val = sigm(val);
    if (EPI == 3) val = gelu_exact(bng * (val - bnm) * bnr + bnb);
    Y[(size_t)row * ldY + col] = val;
  }
}

// ---------------------------------------------------------------------------
// Fused RK4 RHS eval over the 512-dim ODE state (N = K = 512):
//   yt   = Yin + aprev * Kprev                       (folded into A fragment)
//   G    = sigmoid(yt @ GW^T + GB)                   (accumulator 1)
//   U    = yt @ WT^T            (WT = ode_w^T)       (accumulator 2)
//   kv   = gelu(U) / (1 + G)
//   FINAL=0:  Kout = kv
//   FINAL=1:  Yout = Yin + dt6*(K1 + 2*K2 + 2*K3 + kv)   (ping-pong buffer)
// ---------------------------------------------------------------------------
template <bool HAS_PREV, bool FINAL>
__global__ __launch_bounds__(256)
void ode_eval_kernel(const float* __restrict__ Yin, const float* __restrict__ Kprev,
                     float aprev,
                     const float* __restrict__ GW, const float* __restrict__ GB,
                     const float* __restrict__ WT,
                     float* __restrict__ Kout,
                     const float* __restrict__ K1, const float* __restrict__ K2,
                     const float* __restrict__ K3,
                     float* __restrict__ Yout, float dt6) {
  const int lane = threadIdx.x & 31;
  const int wave = threadIdx.x >> 5;
  const int l    = lane & 15;
  const int hf   = lane >> 4;
  const int r0   = blockIdx.y * 16;
  const int c0   = blockIdx.x * 128 + wave * 16;
  const int K    = 512;

  const float* yp = Yin + (size_t)(r0 + l) * K + 4 * hf;
  const float* kp = HAS_PREV ? (Kprev + (size_t)(r0 + l) * K + 4 * hf) : nullptr;
  const float* gp = GW + (size_t)(c0 + l) * K + 4 * hf;
  const float* up = WT + (size_t)(c0 + l) * K + 4 * hf;

  v8f accG = {}, accU = {};
  for (int k = 0; k < K; k += 8) {
    float4 a = *(const float4*)(yp + k);
    if (HAS_PREV) {
      const float4 pk = *(const float4*)(kp + k);
      a.x += aprev * pk.x; a.y += aprev * pk.y;
      a.z += aprev * pk.z; a.w += aprev * pk.w;
    }
    const float4 g = *(const float4*)(gp + k);
    const float4 u = *(const float4*)(up + k);
    v2f av, bv;
    av.x = a.x; av.y = a.y;
    bv.x = g.x; bv.y = g.y;
    accG = WMMA_F32(accG, av, bv);
    bv.x = u.x; bv.y = u.y;
    accU = WMMA_F32(accU, av, bv);
    av.x = a.z; av.y = a.w;
    bv.x = g.z; bv.y = g.w;
    accG = WMMA_F32(accG, av, bv);
    bv.x = u.z; bv.y = u.w;
    accU = WMMA_F32(accU, av, bv);
  }

  const int col = c0 + l;
  const float gbias = GB[col];
#pragma unroll
  for (int v = 0; v < 8; ++v) {
    const int row = r0 + v + 8 * hf;
    const float gs = sigm(accG[v] + gbias);
    const float kv = gelu_exact(accU[v]) / (1.0f + gs);
    const size_t idx = (size_t)row * K + col;
    if (FINAL) {
      Yout[idx] = Yin[idx] + dt6 * (K1[idx] + 2.0f * K2[idx] + 2.0f * K3[idx] + kv);
    } else {
      Kout[idx] = kv;
    }
  }
}

// ---------------------------------------------------------------------------
// Row LayerNorm with optional residual:  Y = LN(A + Res) * g + be
// one block (256 threads) per row; ld applies to A, Res, Y.
// ---------------------------------------------------------------------------
__global__ __launch_bounds__(256)
void ln_kernel(const float* __restrict__ A, const float* __restrict__ Res,
               const float* __restrict__ g, const float* __restrict__ be,
               float* __restrict__ Y, int N, int ld) {
  const int row = blockIdx.x;
  const int tid = threadIdx.x;
  __shared__ float s1[256];
  __shared__ float s2[256];
  const size_t base = (size_t)row * ld;

  float sum = 0.0f, sq = 0.0f;
  for (int i = tid; i < N; i += 256) {
    float v = A[base + i] + (Res ? Res[base + i] : 0.0f);
    sum += v; sq += v * v;
  }
  s1[tid] = sum; s2[tid] = sq;
  __syncthreads();
  for (int s = 128; s > 0; s >>= 1) {
    if (tid < s) { s1[tid] += s1[tid + s]; s2[tid] += s2[tid + s]; }
    __syncthreads();
  }
  const float mean = s1[0] / (float)N;
  const float var  = s2[0] / (float)N - mean * mean;
  const float rstd = rsqrtf(var + LN_EPS);
  for (int i = tid; i < N; i += 256) {
    float v = A[base + i] + (Res ? Res[base + i] : 0.0f);
    Y[base + i] = (v - mean) * rstd * g[i] + be[i];
  }
}

// odeT[n*512 + k] = ode_w[k*512 + n]
__global__ __launch_bounds__(256)
void transpose512_kernel(const float* __restrict__ Win, float* __restrict__ Wout) {
  int i = blockIdx.x * blockDim.x + threadIdx.x;
  if (i >= 512 * 512) return;
  int k = i >> 9, n = i & 511;
  Wout[(size_t)n * 512 + k] = Win[(size_t)k * 512 + n];
}

// logits = z @ wc2^T + bc2 ; softmax ; one block (32 threads) per sample
__global__ __launch_bounds__(32)
void head_kernel(const float* __restrict__ Z, const float* __restrict__ Wc,
                 const float* __restrict__ bc, float* __restrict__ out_logits,
                 float* __restrict__ out_probs) {
  const int row = blockIdx.x;
  const int t   = threadIdx.x;
  __shared__ float lg[10];
  if (t < 10) {
    const float* z = Z + (size_t)row * 256;
    const float* w = Wc + (size_t)t * 256;
    float s = bc[t];
    for (int k = 0; k < 256; ++k) s += z[k] * w[k];
    lg[t] = s;
    out_logits[row * 10 + t] = s;
  }
  __syncthreads();
  if (t < 10) {
    float m = lg[0];
    for (int j = 1; j < 10; ++j) m = fmaxf(m, lg[j]);
    float den = 0.0f;
    for (int j = 0; j < 10; ++j) den += expf(lg[j] - m);
    out_probs[row * 10 + t] = expf(lg[t] - m) / den;
  }
}

// ---------------------------------------------------------------------------
// Host orchestration
// ---------------------------------------------------------------------------
static inline void launch_gemm(int epi, const float* X, const float* W,
                               const float* b, float* Y, int K, int N, int ldY,
                               hipStream_t s,
                               const float* bn_g = nullptr, const float* bn_b = nullptr,
                               const float* bn_m = nullptr, const float* bn_v = nullptr) {
  dim3 grid((N + 127) / 128, NROWS / 16), block(256);
  switch (epi) {
    case 0: gemm_xwt_kernel<0><<<grid, block, 0, s>>>(X, W, b, Y, K, N, ldY,
                bn_g, bn_b, bn_m, bn_v); break;
    case 1: gemm_xwt_kernel<1><<<grid, block, 0, s>>>(X, W, b, Y, K, N, ldY,
                bn_g, bn_b, bn_m, bn_v); break;
    case 2: gemm_xwt_kernel<2><<<grid, block, 0, s>>>(X, W, b, Y, K, N, ldY,
                bn_g, bn_b, bn_m, bn_v); break;
    default: gemm_xwt_kernel<3><<<grid, block, 0, s>>>(X, W, b, Y, K, N, ldY,
                bn_g, bn_b, bn_m, bn_v); break;
  }
}

static inline void launch_ln(const float* A, const float* Res, const float* g,
                             const float* be, float* Y, int N, int ld,
                             hipStream_t s) {
  ln_kernel<<<NROWS, 256, 0, s>>>(A, Res, g, be, Y, N, ld);
}

extern "C" void kernel_launch(void* const* d_in, const int* in_sizes, int n_in,
                              void* d_out, int out_size, void* d_ws, size_t ws_size,
                              hipStream_t stream) {
  (void)in_sizes; (void)n_in; (void)out_size; (void)ws_size;

  // inputs in setup_inputs() dict order
  const float* x    = (const float*)d_in[0];   // [256, 12288]
  const float* w1   = (const float*)d_in[1];
  const float* b1   = (const float*)d_in[2];
  const float* g1   = (const float*)d_in[3];
  const float* be1  = (const float*)d_in[4];
  const float* bw   = (const float*)d_in[5];   // [4,256,2048]
  const float* bb   = (const float*)d_in[6];   // [4,256]
  const float* bg   = (const float*)d_in[7];
  const float* bbe  = (const float*)d_in[8];
  const float* fw   = (const float*)d_in[9];   // [2048,1024]
  const float* fb   = (const float*)d_in[10];
  const float* bn_g = (const float*)d_in[11];
  const float* bn_b = (const float*)d_in[12];
  const float* bn_m = (const float*)d_in[13];
  const float* bn_v = (const float*)d_in[14];
  const float* wqkv = (const float*)d_in[15];  // [6144,2048]
  const float* bqkv = (const float*)d_in[16];
  const float* wo   = (const float*)d_in[17];
  const float* bo   = (const float*)d_in[18];
  const float* ng   = (const float*)d_in[19];
  const float* nbe  = (const float*)d_in[20];
  const float* w2   = (const float*)d_in[21];
  const float* b2   = (const float*)d_in[22];
  const float* g2   = (const float*)d_in[23];
  const float* be2  = (const float*)d_in[24];
  const float* w3   = (const float*)d_in[25];
  const float* b3   = (const float*)d_in[26];
  const float* g3   = (const float*)d_in[27];
  const float* be3  = (const float*)d_in[28];
  const float* gw   = (const float*)d_in[29];  // [512,512]
  const float* gb   = (const float*)d_in[30];
  const float* odew = (const float*)d_in[31];  // [512,512] (used as y @ W)
  const float* wc1  = (const float*)d_in[32];
  const float* bc1  = (const float*)d_in[33];
  const float* gc   = (const float*)d_in[34];
  const float* bec  = (const float*)d_in[35];
  const float* wc2  = (const float*)d_in[36];
  const float* bc2  = (const float*)d_in[37];

  float* ws = (float*)d_ws;
  size_t off = 0;
  float* H    = ws + off; off += (size_t)NROWS * 2048;
  float* CAT  = ws + off; off += (size_t)NROWS * 1024;
  float* F    = ws + off; off += (size_t)NROWS * 2048;
  float* T1   = ws + off; off += (size_t)NROWS * 2048;
  float* Abuf = ws + off; off += (size_t)NROWS * 2048;
  float* H2   = ws + off; off += (size_t)NROWS * 1024;
  float* Y0   = ws + off; off += (size_t)NROWS * 512;   // ODE state (ping)
  float* Y1   = ws + off; off += (size_t)NROWS * 512;   // ODE state (pong)
  float* K1   = ws + off; off += (size_t)NROWS * 512;
  float* K2   = ws + off; off += (size_t)NROWS * 512;
  float* K3   = ws + off; off += (size_t)NROWS * 512;
  float* ODET = ws + off; off += (size_t)512 * 512;
  float* Z    = ws + off; off += (size_t)NROWS * 256;

  // ---- feature net: h = LN(gelu(x @ w1^T + b1)) -------------------------
  launch_gemm(1, x, w1, b1, H, 12288, 2048, 2048, stream);
  launch_ln(H, nullptr, g1, be1, H, 2048, 2048, stream);

  // ---- MultiScaleLiquid: 4 branches -> CAT [256,1024] -------------------
  for (int i = 0; i < 4; ++i) {
    launch_gemm(1, H, bw + (size_t)i * 256 * 2048, bb + i * 256,
                CAT + i * 256, 2048, 256, 1024, stream);
    launch_ln(CAT + i * 256, nullptr, bg + i * 256, bbe + i * 256,
              CAT + i * 256, 256, 1024, stream);
  }

  // ---- fusion GEMM with fused BatchNorm(eval)+GELU epilogue ------------
  launch_gemm(3, CAT, fw, fb, F, 1024, 2048, 2048, stream,
              bn_g, bn_b, bn_m, bn_v);

  // ---- degenerate attention: attn = (f @ wv^T + bv) @ wo^T + bo --------
  launch_gemm(0, F, wqkv + (size_t)2 * 2048 * 2048, bqkv + 2 * 2048,
              T1, 2048, 2048, 2048, stream);
  launch_gemm(0, T1, wo, bo, Abuf, 2048, 2048, 2048, stream);
  launch_ln(H, Abuf, ng, nbe, H, 2048, 2048, stream);   // h = LN(h + attn)

  // ---- MLP down to 512 --------------------------------------------------
  launch_gemm(1, H, w2, b2, H2, 2048, 1024, 1024, stream);
  launch_ln(H2, nullptr, g2, be2, H2, 1024, 1024, stream);
  launch_gemm(1, H2, w3, b3, Y0, 1024, 512, 512, stream);
  launch_ln(Y0, nullptr, g3, be3, Y0, 512, 512, stream);

  // ---- ODE: fixed-step RK4, fully fused RHS (4 kernels / step) ---------
  transpose512_kernel<<<(512 * 512 + 255) / 256, 256, 0, stream>>>(odew, ODET);
  const float dt = 3.0f / (float)ODE_STEPS;
  const dim3 og(4, NROWS / 16), ob(256);
  float* Ya = Y0;
  float* Yb = Y1;
  for (int s = 0; s < ODE_STEPS; ++s) {
    ode_eval_kernel<false, false><<<og, ob, 0, stream>>>(
        Ya, nullptr, 0.0f, gw, gb, ODET, K1, nullptr, nullptr, nullptr,
        nullptr, 0.0f);
    ode_eval_kernel<true, false><<<og, ob, 0, stream>>>(
        Ya, K1, 0.5f * dt, gw, gb, ODET, K2, nullptr, nullptr, nullptr,
        nullptr, 0.0f);
    ode_eval_kernel<true, false><<<og, ob, 0, stream>>>(
        Ya, K2, 0.5f * dt, gw, gb, ODET, K3, nullptr, nullptr, nullptr,
        nullptr, 0.0f);
    ode_eval_kernel<true, true><<<og, ob, 0, stream>>>(
        Ya, K3, dt, gw, gb, ODET, nullptr, K1, K2, K3, Yb, dt / 6.0f);
    float* tmp = Ya; Ya = Yb; Yb = tmp;   // ping-pong state buffer
  }

  // ---- classifier -------------------------------------------------------
  launch_gemm(1, Ya, wc1, bc1, Z, 512, 256, 256, stream);
  launch_ln(Z, nullptr, gc, bec, Z, 256, 256, stream);

  float* out_logits = (float*)d_out;
  float* out_probs  = out_logits + NROWS * 10;
  head_kernel<<<NROWS, 32, 0, stream>>>(Z, wc2, bc2, out_logits, out_probs);
}